// PhaseChannelizer_58299886076100
// MI455X (gfx1250) — compile-verified
//
#include <hip/hip_runtime.h>
#include <cmath>
#include <stdint.h>

typedef __attribute__((ext_vector_type(2))) float v2f;
typedef __attribute__((ext_vector_type(8))) float v8f;
typedef __attribute__((ext_vector_type(4))) unsigned int u32x4;
typedef __attribute__((ext_vector_type(8))) int i32x8;
typedef __attribute__((ext_vector_type(4))) int i32x4;

#define WINDOW  15
#define EPSF    1e-6f
#define TILE_T  16
#define KDIM    32        // time rows staged per tile: j in [i0-15, i0+16]
#define CHUNK_M 256       // channels per workgroup
#define LDSW    264       // padded LDS row stride (floats): 2*264 % 64 = 16 -> conflict-free B reads

__device__ __forceinline__ float bump_w(int d) {
    // d in [0, WINDOW]; matches reference: clip is identity here
    float u = fminf((float)d * (1.0f / (float)WINDOW), 1.0f - EPSF);
    return expf(1.0f - 1.0f / (1.0f - u * u + EPSF));
}

__global__ __launch_bounds__(256)
void bump_conv_wmma(const float* __restrict__ x,
                    const float* __restrict__ mask,
                    const float* __restrict__ gate_raw,
                    float* __restrict__ out,
                    int T, int M) {
    __shared__ float Xs[KDIM * LDSW];

    const int m0 = blockIdx.x * CHUNK_M;
    const int i0 = blockIdx.y * TILE_T;
    const int b  = blockIdx.z;
    const int j0 = i0 - WINDOW;                 // first (possibly negative) window row
    const int t  = threadIdx.x;
    const size_t xbase = ((size_t)b * T) * M;

    // ---- stage x[b, j0 .. j0+31, m0 .. m0+255] into LDS via the Tensor Data Mover ----
    const int jstart = (j0 < 0) ? 0 : j0;
    const int skip   = jstart - j0;             // leading rows with j<0 (first tile only)

    if (skip > 0) {                             // zero-fill rows the DMA won't touch
        for (int idx = t; idx < skip * LDSW; idx += 256) Xs[idx] = 0.0f;
    }

    if (t < 32) {                               // one wave issues the 2D tile descriptor
        const float* gptr = x + xbase + (size_t)jstart * M + m0;
        unsigned long long ga = (unsigned long long)(uintptr_t)gptr;
        unsigned lds = (unsigned)(uintptr_t)(&Xs[skip * LDSW]);
        unsigned rows_av   = (unsigned)(T - jstart);   // tensor_dim1: rows past this clamp to 0
        unsigned rows_tile = (unsigned)(KDIM - skip);  // tile_dim1

        u32x4 g0;
        g0[0] = 1u;                                         // count=1, user descriptor
        g0[1] = lds;                                        // LDS byte address
        g0[2] = (unsigned)(ga & 0xFFFFFFFFull);             // global_addr[31:0]
        g0[3] = (unsigned)((ga >> 32) & 0x1FFFFFFull)       // global_addr[56:32]
              | (2u << 30);                                 // type=2 (image/D#)

        i32x8 g1;
        g1[0] = (2 << 16)      // data_size = 4B
              | (1 << 20)      // pad_enable
              | (7 << 22)      // pad_interval: every 256 dwords (one row)
              | (7 << 25);     // pad_amount: 8 dwords -> LDS row stride 264 floats
        g1[1] = (int)((CHUNK_M & 0xFFFF) << 16);            // tensor_dim0[15:0] (=256)
        g1[2] = (int)((rows_av & 0xFFFFu) << 16);           // tensor_dim0[31:16]=0 | tensor_dim1[15:0]
        g1[3] = (int)(((rows_av >> 16) & 0xFFFFu)           // tensor_dim1[31:16]
              | ((unsigned)CHUNK_M << 16));                 // tile_dim0 = 256
        g1[4] = (int)(rows_tile & 0xFFFFu);                 // tile_dim1 | tile_dim2=0
        g1[5] = M;                                          // tensor_dim0_stride (elements)
        g1[6] = 0;                                          // stride hi | dim1_stride lo
        g1[7] = 0;                                          // dim1_stride hi

        i32x4 gz = {0, 0, 0, 0};
#if defined(__clang_major__) && __clang_major__ >= 23
        i32x8 gz8 = {0, 0, 0, 0, 0, 0, 0, 0};
        __builtin_amdgcn_tensor_load_to_lds(g0, g1, gz, gz, gz8, 0);
#else
        __builtin_amdgcn_tensor_load_to_lds(g0, g1, gz, gz, 0);
#endif
        __builtin_amdgcn_s_wait_tensorcnt(0);
    }
    __syncthreads();

    // ---- per-lane A-matrix (banded Toeplitz, masked + row-normalized) ----
    const int lane = t & 31;
    const int half = lane >> 4;          // 0: lanes 0-15, 1: lanes 16-31
    const int r    = lane & 15;          // output row within time tile
    const int i    = i0 + r;             // global time index

    float wts[WINDOW + 1];
    float S = 0.0f;
    #pragma unroll
    for (int d = 0; d <= WINDOW; ++d) {
        int j = i - d;
        float w = (j >= 0) ? bump_w(d) * mask[(size_t)i * T + j] : 0.0f;
        wts[d] = w;
        S += w;
    }
    const float invS = 1.0f / fmaxf(S, EPSF);

    // f32 16x16x4 A layout: element K = 4*s + 2*half + comp lives at lane row r
    v2f afrag[8];
    #pragma unroll
    for (int s = 0; s < 8; ++s) {
        #pragma unroll
        for (int comp = 0; comp < 2; ++comp) {
            int k = 4 * s + 2 * half + comp;   // column of the 16x32 A tile
            int d = r + WINDOW - k;            // i - j
            afrag[s][comp] = (d >= 0 && d <= WINDOW) ? wts[d] * invS : 0.0f;
        }
    }

    // ---- per-wave 16x16 output tiles: out_tile = A(16x32) x Xs(32x16) ----
    const int wave = t >> 5;                   // 0..7
    #pragma unroll
    for (int ct = 0; ct < 2; ++ct) {
        const int chan = wave * 32 + ct * 16 + r;   // this lane's channel column
        v8f acc = {};
        #pragma unroll
        for (int s = 0; s < 8; ++s) {
            // f32 4x16 B layout mirrors A: K = 4*s + 2*half + vgpr, N = lane&15
            int rowb = 4 * s + 2 * half;
            v2f bfrag;
            bfrag[0] = Xs[(rowb    ) * LDSW + chan];
            bfrag[1] = Xs[(rowb + 1) * LDSW + chan];
            acc = __builtin_amdgcn_wmma_f32_16x16x4_f32(
                false, afrag[s], false, bfrag, (short)0, acc, false, false);
        }
        // softplus gate for this lane's channel (stable form)
        float g    = gate_raw[m0 + chan];
        float gate = fmaxf(g, 0.0f) + log1pf(expf(-fabsf(g)));
        // D layout: VGPR v holds row M = v + 8*half, col N = lane&15
        #pragma unroll
        for (int v = 0; v < 8; ++v) {
            int orow = v + 8 * half;
            out[((size_t)b * T + (i0 + orow)) * M + m0 + chan] = acc[v] * gate;
        }
    }
}

extern "C" void kernel_launch(void* const* d_in, const int* in_sizes, int n_in,
                              void* d_out, int out_size, void* d_ws, size_t ws_size,
                              hipStream_t stream) {
    (void)n_in; (void)out_size; (void)d_ws; (void)ws_size;
    const float* x        = (const float*)d_in[0];
    const float* mask     = (const float*)d_in[1];
    const float* gate_raw = (const float*)d_in[2];
    float*       out      = (float*)d_out;

    const int M = in_sizes[2];                               // gate_raw: (M,)
    const int T = (int)(sqrt((double)in_sizes[1]) + 0.5);    // mask: (T,T)
    const int B = in_sizes[0] / (T * M);                     // x: (B,T,M)

    dim3 block(256, 1, 1);
    dim3 grid((unsigned)((M + CHUNK_M - 1) / CHUNK_M),
              (unsigned)((T + TILE_T - 1) / TILE_T),
              (unsigned)B);
    bump_conv_wmma<<<grid, block, 0, stream>>>(x, mask, gate_raw, out, T, M);
}